// GraphNet_62294205661623
// MI455X (gfx1250) — compile-verified
//
#include <hip/hip_runtime.h>
#include <math.h>

typedef __attribute__((ext_vector_type(2))) float v2f;
typedef __attribute__((ext_vector_type(4))) float v4f;
typedef __attribute__((ext_vector_type(8))) float v8f;

// ---- problem constants (fixed by the reference) ----
static constexpr int CONT   = 49992;
static constexpr int NCAT   = 8;
static constexpr int CATL   = 16;
static constexpr int SDIM   = 448;
static constexpr int NNODE  = 50000;     // CONT + NCAT
static constexpr int HIDDEN = 128;
static constexpr int NCLS   = 10;
static constexpr int BATCH  = 1024;
static constexpr int KDIM   = 448;       // effective K (padded cols are zero)
static constexpr int KCHUNK = 112;       // 4 chunks of 112 = 448
static constexpr int MTILES = NNODE / 16;  // 3125 exactly
static constexpr float BETA = 1.1f;

// ---- workspace layout (float offsets) ----
static constexpr long OFF_CAT  = 0;          // 8*448 = 3584
static constexpr long OFF_DEG  = 4096;       // 50000 (deg, then dinv in-place)
static constexpr long OFF_GSUM = 56320;      // 128
static constexpr long OFF_GDOT = 56448;      // 1
static constexpr long OFF_XW   = 57344;      // 6,400,000
static constexpr long OFF_H    = 6457344;    // 6,400,000
// total ~12.86M floats = 51.4 MB

// ------------------------------------------------------------------
__global__ void zero_kernel(v4f* __restrict__ p, long n4) {
  long i = (long)blockIdx.x * blockDim.x + threadIdx.x;
  long stride = (long)gridDim.x * blockDim.x;
  v4f z = {};
  for (; i < n4; i += stride) p[i] = z;
}

// cat_rows[c][s] = sum_k cat_x[c][s][k] * emb_w[c][k]
__global__ void embed_kernel(const float* __restrict__ cat_x,
                             const float* __restrict__ emb_w,
                             float* __restrict__ cat_rows) {
  int idx = blockIdx.x * blockDim.x + threadIdx.x;
  if (idx >= NCAT * SDIM) return;
  int c = idx / SDIM, s = idx % SDIM;
  const float* cx = cat_x + ((long)c * SDIM + s) * CATL;
  const float* ew = emb_w + (long)c * CATL;
  float acc = 0.0f;
  #pragma unroll
  for (int k = 0; k < CATL; ++k) acc = fmaf(cx[k], ew[k], acc);
  cat_rows[idx] = acc;
}

// deg[col[e]] += ew[e]
__global__ void degree_kernel(const int* __restrict__ ei,
                              const float* __restrict__ ew,
                              float* __restrict__ deg, int E) {
  int e = blockIdx.x * blockDim.x + threadIdx.x;
  if (e >= E) return;
  atomicAdd(deg + ei[E + e], ew[e]);
}

// dinv[n] = rsqrt(deg[n] + 1)   (self-loop weight 1; always > 0)
__global__ void dinv_kernel(float* __restrict__ deg) {
  int n = blockIdx.x * blockDim.x + threadIdx.x;
  if (n >= NNODE) return;
  deg[n] = rsqrtf(deg[n] + 1.0f);
}

// ------------------------------------------------------------------
// xw[50000 x 128] = x[50000 x 448] @ conv_w[448 x 128] via f32 WMMA.
// 8 waves / block; each wave: one 16-row M-tile x all 8 N-tiles.
//
// LDS layout for the 112x128 conv_w chunk: K-pairs interleaved so each
// B fragment is ONE aligned ds_load_b64, with a 32-dword rotation per
// K-pair row so lanes 0-15 (K,K+1) and lanes 16-31 (K+2,K+3) hit
// disjoint halves of the 64 LDS banks (conflict-free full-wave b64).
//   element (k, n) -> lds[(k>>1)*256 + ((2n + ((k>>1 & 1)<<5)) & 255) + (k&1)]
__global__ __launch_bounds__(256) void gemm_xw_kernel(
    const float* __restrict__ num_x, const float* __restrict__ cat_rows,
    const float* __restrict__ conv_w, float* __restrict__ xw) {
  __shared__ float ldsB[(KCHUNK / 2) * 256];   // 57,344 bytes

  const int tid   = threadIdx.x;
  const int lane  = tid & 31;
  const int wave  = tid >> 5;
  const int mtile = blockIdx.x * 8 + wave;
  const bool active = (mtile < MTILES);
  const int l15   = lane & 15;
  const int koff  = (lane >> 4) << 1;  // lanes 16-31 hold K+2,K+3
  const int tw    = l15 << 1;          // 2*n within a K-pair row

  // per-lane A row pointer (row < CONT -> num_x, else cat_rows)
  const float* xptr = nullptr;
  if (active) {
    int row = mtile * 16 + l15;
    xptr = (row < CONT) ? (num_x + (long)row * SDIM)
                        : (cat_rows + (long)(row - CONT) * SDIM);
    xptr += koff;
  }

  v8f acc[8] = {};

  for (int kc = 0; kc < KDIM / KCHUNK; ++kc) {
    const int kb = kc * KCHUNK;
    __syncthreads();
    for (int idx = tid; idx < KCHUNK * HIDDEN; idx += 256) {
      int k = idx >> 7, n = idx & 127;
      int p = k >> 1;
      int col = ((2 * n + ((p & 1) << 5)) & 255) + (k & 1);
      ldsB[p * 256 + col] = conv_w[(long)kb * HIDDEN + idx];
    }
    __syncthreads();

    if (active) {
      for (int kk = 0; kk < KCHUNK; kk += 4) {
        if ((kk & 31) == 0)  // prefetch the A stream one K-chunk ahead
          __builtin_prefetch(xptr + kb + kk + KCHUNK, 0, 1);
        // A fragment: lane l -> row (l&15), K = kk + koff + {0,1}
        v2f a = *(const v2f*)(xptr + kb + kk);
        const int p   = (kk + koff) >> 1;
        const int rb  = p * 256;
        const int rot = (p & 1) << 5;
        #pragma unroll
        for (int t = 0; t < 8; ++t) {
          // B fragment: single aligned b64 = {B[k][n], B[k+1][n]}
          v2f b = *(const v2f*)&ldsB[rb + ((rot + (t << 5) + tw) & 255)];
          acc[t] = __builtin_amdgcn_wmma_f32_16x16x4_f32(
              false, a, false, b, (short)0, acc[t], false, false);
        }
      }
    }
  }

  if (active) {
    // C/D layout: VGPR r, lane l -> M = r + 8*(l>>4), N = l&15
    const int rbase = mtile * 16 + ((lane >> 4) << 3);
    #pragma unroll
    for (int t = 0; t < 8; ++t)
      #pragma unroll
      for (int r = 0; r < 8; ++r)
        xw[(long)(rbase + r) * HIDDEN + t * 16 + l15] = acc[t][r];
  }
}

// ------------------------------------------------------------------
// h[col] += norm * xw[row] for each edge; one wave per edge, 4 floats/lane.
// xw (25.6 MB) and h (25.6 MB) are L2-resident (192 MB L2).
__global__ __launch_bounds__(256) void edge_agg_kernel(
    const int* __restrict__ ei, const float* __restrict__ ew,
    const float* __restrict__ dinv, const float* __restrict__ xw,
    float* __restrict__ h, int E) {
  int e = blockIdx.x * 8 + (threadIdx.x >> 5);
  if (e >= E) return;
  int lane = threadIdx.x & 31;
  int row = ei[e];
  int col = ei[E + e];
  float norm = dinv[row] * ew[e] * dinv[col];
  v4f src = *(const v4f*)(xw + (long)row * HIDDEN + lane * 4);
  float* dst = h + (long)col * HIDDEN + lane * 4;
  atomicAdd(dst + 0, src.x * norm);
  atomicAdd(dst + 1, src.y * norm);
  atomicAdd(dst + 2, src.z * norm);
  atomicAdd(dst + 3, src.w * norm);
}

// ------------------------------------------------------------------
// gsum[j] += sum_n relu(h[n][j] + dinv[n]^2 * xw[n][j])  (self-loop folded in)
static constexpr int POOL_NODES = 250;  // 200 blocks * 250 = 50000
__global__ __launch_bounds__(128) void pool_kernel(
    const float* __restrict__ h, const float* __restrict__ xw,
    const float* __restrict__ dinv, float* __restrict__ gsum) {
  int j = threadIdx.x;
  int n0 = blockIdx.x * POOL_NODES;
  float sum = 0.0f;
  for (int i = 0; i < POOL_NODES; ++i) {
    int n = n0 + i;
    float di = dinv[n];
    float v = h[(long)n * HIDDEN + j] + di * di * xw[(long)n * HIDDEN + j];
    sum += fmaxf(v, 0.0f);
  }
  atomicAdd(gsum + j, sum);
}

// gdot = sum_j (gsum[j]/N) * fc_w[NCLS + j] + fc_b
__global__ __launch_bounds__(128) void gdot_kernel(
    const float* __restrict__ gsum, const float* __restrict__ fc_w,
    const float* __restrict__ fc_b, float* __restrict__ gdot) {
  __shared__ float red[128];
  int j = threadIdx.x;
  red[j] = (gsum[j] * (1.0f / (float)NNODE)) * fc_w[NCLS + j];
  __syncthreads();
  for (int s = 64; s > 0; s >>= 1) {
    if (j < s) red[j] += red[j + s];
    __syncthreads();
  }
  if (j == 0) gdot[0] = red[0] + fc_b[0];
}

// out[b][c] = vanilla[b][c] / (softplus(BETA * t_lin[b]) / BETA)
__global__ __launch_bounds__(256) void final_kernel(
    const float* __restrict__ vanilla, const float* __restrict__ fc_w,
    const float* __restrict__ gdot, float* __restrict__ out) {
  int b = blockIdx.x * blockDim.x + threadIdx.x;
  if (b >= BATCH) return;
  const float* vrow = vanilla + (long)b * NCLS;
  float s = gdot[0];
  #pragma unroll
  for (int c = 0; c < NCLS; ++c) s = fmaf(vrow[c], fc_w[c], s);
  float x = BETA * s;
  // stable softplus: max(x,0) + log1p(exp(-|x|))
  float sp = fmaxf(x, 0.0f) + log1pf(__expf(-fabsf(x)));
  float t = sp / BETA;
  float inv_t = 1.0f / t;
  #pragma unroll
  for (int c = 0; c < NCLS; ++c) out[(long)b * NCLS + c] = vrow[c] * inv_t;
}

// ------------------------------------------------------------------
extern "C" void kernel_launch(void* const* d_in, const int* in_sizes, int n_in,
                              void* d_out, int out_size, void* d_ws, size_t ws_size,
                              hipStream_t stream) {
  const float* num_x   = (const float*)d_in[0];
  const float* cat_x   = (const float*)d_in[1];
  const float* edge_w  = (const float*)d_in[2];
  const float* vanilla = (const float*)d_in[3];
  const float* emb_w   = (const float*)d_in[4];
  const float* conv_w  = (const float*)d_in[5];
  const float* fc_w    = (const float*)d_in[6];
  const float* fc_b    = (const float*)d_in[7];
  const int*   ei      = (const int*)d_in[8];
  float* out = (float*)d_out;
  float* ws  = (float*)d_ws;

  const int E = in_sizes[2];

  float* cat_rows = ws + OFF_CAT;
  float* deg      = ws + OFF_DEG;   // becomes dinv in-place
  float* gsum     = ws + OFF_GSUM;
  float* gdot     = ws + OFF_GDOT;
  float* xw       = ws + OFF_XW;
  float* h        = ws + OFF_H;

  // zero the accumulators (must be re-zeroed every call)
  zero_kernel<<<1024, 256, 0, stream>>>((v4f*)h, (long)NNODE * HIDDEN / 4);
  zero_kernel<<<64, 256, 0, stream>>>((v4f*)deg, (long)NNODE / 4);
  zero_kernel<<<1, 32, 0, stream>>>((v4f*)gsum, (long)HIDDEN / 4);

  embed_kernel<<<(NCAT * SDIM + 255) / 256, 256, 0, stream>>>(cat_x, emb_w, cat_rows);
  degree_kernel<<<(E + 255) / 256, 256, 0, stream>>>(ei, edge_w, deg, E);
  dinv_kernel<<<(NNODE + 255) / 256, 256, 0, stream>>>(deg);

  gemm_xw_kernel<<<(MTILES + 7) / 8, 256, 0, stream>>>(num_x, cat_rows, conv_w, xw);

  edge_agg_kernel<<<(E + 7) / 8, 256, 0, stream>>>(ei, edge_w, deg, xw, h, E);

  pool_kernel<<<NNODE / POOL_NODES, 128, 0, stream>>>(h, xw, deg, gsum);
  gdot_kernel<<<1, 128, 0, stream>>>(gsum, fc_w, fc_b, gdot);
  final_kernel<<<(BATCH + 255) / 256, 256, 0, stream>>>(vanilla, fc_w, gdot, out);
}